// SpikingNeuron_27736898798059
// MI455X (gfx1250) — compile-verified
//
#include <hip/hip_runtime.h>
#include <stdint.h>

// LIF scan: spikes[n,t] = (beta*mem + x >= thr), mems[n,t] = mem-before-step.
// Pure streaming: 384 MB @ 23.3 TB/s ~= 16.5 us floor. wmma N/A (nonlinear reset).
// CDNA5 paths: GLOBAL_LOAD_ASYNC_TO_LDS_B128 (ASYNCcnt) staging, NT cache hints,
// s_wait_asynccnt, coalesced B128 stores via LDS transpose tiles.

#define LIF_BLOCK 256          // neurons per block (8 wave32)
#define LIF_TC    32           // time-steps per tile (128 B per neuron-row chunk)
#define LIF_LSTR  36           // padded LDS row stride in floats (144 B, 16B-aligned rows)

typedef float v4f __attribute__((ext_vector_type(4)));  // native vector: b128 + NT-capable

__device__ __forceinline__ uint32_t lif_lds_off(const void* p) {
  // Generic pointer to LDS on AMDGPU: low 32 bits are the LDS byte offset.
  return (uint32_t)(uintptr_t)p;
}

__device__ __forceinline__ void lif_async_load_b128(uint32_t lds_byte_off, const float* g) {
  uint64_t ga = (uint64_t)(uintptr_t)g;
  // GLOBAL_LOAD_ASYNC_TO_LDS_B128: VDST = LDS byte address, VADDR = 64-bit global addr.
  // Non-temporal: x is streamed exactly once; don't pollute L2.
  asm volatile("global_load_async_to_lds_b128 %0, %1, off th:TH_LOAD_NT"
               :
               : "v"(lds_byte_off), "v"(ga)
               : "memory");
}

__device__ __forceinline__ void lif_wait_async0() {
  asm volatile("s_wait_asynccnt 0x0" ::: "memory");
}

__global__ __launch_bounds__(LIF_BLOCK) void lif_scan_kernel(
    const float* __restrict__ x,
    const float* __restrict__ beta_logit,
    const float* __restrict__ threshold,
    const float* __restrict__ init_u,
    float* __restrict__ spikes,
    float* __restrict__ mems,
    int T_)
{
  __shared__ float shx[2][LIF_BLOCK * LIF_LSTR];  // double-buffered x tiles (2 x 36 KB)
  __shared__ float shs[LIF_BLOCK * LIF_LSTR];     // spike tile (36 KB)
  __shared__ float shm[LIF_BLOCK * LIF_LSTR];     // mem-in tile (36 KB)

  const unsigned T   = (unsigned)T_;
  const unsigned tid = threadIdx.x;
  const unsigned n0  = blockIdx.x * LIF_BLOCK;

  const float thr  = threshold[0];
  const float beta = 1.0f / (1.0f + __expf(-beta_logit[0]));
  float mem = init_u[n0 + tid] * thr;

  const unsigned nTiles = T / LIF_TC;

  // Cooperative transfer mapping: tile = 256 x 32 floats = 2048 float4.
  // Thread handles 8 float4 elements: k in [0,8), row = k*32 + (tid>>3), c4 = tid&7.
  // => 8 consecutive lanes move one contiguous 128 B row-chunk (fully coalesced).
  // All offsets fit in u32 (N*T = 33.5M elements); k*32*T is uniform -> SALU.
  const unsigned rbase = tid >> 3;              // 0..31
  const unsigned cbase = (tid & 7u) * 4u;       // float offset within row chunk
  const unsigned goff0 = (n0 + rbase) * T + cbase;       // per-thread global base (floats)
  const unsigned loff0 = rbase * LIF_LSTR + cbase;       // per-thread LDS base (floats)

  // Prologue: async-prefetch tile 0 into buffer 0.
  #pragma unroll
  for (unsigned k = 0; k < LIF_TC / 4; ++k) {
    lif_async_load_b128(lif_lds_off(&shx[0][loff0 + k * (32u * LIF_LSTR)]),
                        x + (goff0 + k * (32u * T)));
  }

  for (unsigned i = 0; i < nTiles; ++i) {
    const unsigned tc  = i * LIF_TC;
    const unsigned cur = i & 1u;

    lif_wait_async0();   // my share of tile i landed in LDS
    __syncthreads();     // everyone's share landed; prev iter's LDS readers done

    // Prefetch tile i+1 into the other buffer; overlaps the compute below.
    if (i + 1 < nTiles) {
      const unsigned nb = cur ^ 1u;
      #pragma unroll
      for (unsigned k = 0; k < LIF_TC / 4; ++k) {
        lif_async_load_b128(lif_lds_off(&shx[nb][loff0 + k * (32u * LIF_LSTR)]),
                            x + (goff0 + (tc + LIF_TC) + k * (32u * T)));
      }
    }

    // Sequential recurrence: 32 steps, 4 per LDS b128 load/store.
    #pragma unroll
    for (unsigned t4 = 0; t4 < LIF_TC / 4; ++t4) {
      const v4f xv = *(const v4f*)&shx[cur][tid * LIF_LSTR + t4 * 4];
      float xs[4] = {xv.x, xv.y, xv.z, xv.w};
      float ss[4], ms[4];
      #pragma unroll
      for (int j = 0; j < 4; ++j) {
        ms[j] = mem;                            // memory entering this step
        const float mu = beta * mem + xs[j];    // state update
        const float h  = (mu - thr) >= 0.0f ? 1.0f : 0.0f;
        ss[j] = h;                              // spike == hard (surrogate cancels fwd)
        mem = mu - h * thr;                     // detached reset
      }
      const v4f sv = {ss[0], ss[1], ss[2], ss[3]};
      const v4f mv = {ms[0], ms[1], ms[2], ms[3]};
      *(v4f*)&shs[tid * LIF_LSTR + t4 * 4] = sv;
      *(v4f*)&shm[tid * LIF_LSTR + t4 * 4] = mv;
    }

    __syncthreads();     // output tiles complete in LDS

    // Coalesced non-temporal B128 stores of both output tiles (written once, never re-read).
    #pragma unroll
    for (unsigned k = 0; k < LIF_TC / 4; ++k) {
      const unsigned lo = loff0 + k * (32u * LIF_LSTR);
      const unsigned go = goff0 + tc + k * (32u * T);
      const v4f s = *(const v4f*)&shs[lo];
      const v4f m = *(const v4f*)&shm[lo];
      __builtin_nontemporal_store(s, (v4f*)&spikes[go]);
      __builtin_nontemporal_store(m, (v4f*)&mems[go]);
    }
    // Top-of-loop barrier protects shs/shm and the swapped x buffer next iteration.
  }
}

extern "C" void kernel_launch(void* const* d_in, const int* in_sizes, int n_in,
                              void* d_out, int out_size, void* d_ws, size_t ws_size,
                              hipStream_t stream) {
  const float* x     = (const float*)d_in[0];   // [N, T]
  const float* beta_ = (const float*)d_in[1];   // [1]
  const float* thr   = (const float*)d_in[2];   // [1]
  const float* iu    = (const float*)d_in[3];   // [N]

  const int N = in_sizes[3];
  const int T = in_sizes[0] / N;

  float* spikes = (float*)d_out;                 // [N, T]
  float* mems   = spikes + (size_t)N * T;        // [N, T]

  lif_scan_kernel<<<dim3(N / LIF_BLOCK), dim3(LIF_BLOCK), 0, stream>>>(
      x, beta_, thr, iu, spikes, mems, T);
}